// MHCLayer_20555713478899
// MI455X (gfx1250) — compile-verified
//
#include <hip/hip_runtime.h>
#include <hip/hip_bf16.h>
#include <stdint.h>

// Shapes from the reference: B=8192, N=4, C=4096 (B derived from in_sizes at launch).
#define NSTR    4
#define CDIM    4096
#define TPAD    16
#define TSTR    (CDIM + TPAD)      // padded LDS row stride (floats)
#define THREADS 256
#define EPSF    1e-6f

typedef float v2f __attribute__((ext_vector_type(2)));
typedef float v4f __attribute__((ext_vector_type(4)));
typedef float v8f __attribute__((ext_vector_type(8)));

// Round-to-nearest-even fp32 -> bf16 -> fp32 round trip (matches jnp bf16 cast).
__device__ __forceinline__ float bf16_roundtrip(float v) {
    uint32_t u = __float_as_uint(v);
    uint32_t r = (u + 0x7FFFu + ((u >> 16) & 1u)) & 0xFFFF0000u;
    return __uint_as_float(r);
}

__global__ __launch_bounds__(THREADS) void mhc_layer_kernel(
    const float* __restrict__ x,       // [B, 4, 4096]
    const float* __restrict__ w,       // [4096]
    const float* __restrict__ Hpre,    // [4]
    const float* __restrict__ Hpost,   // [4]
    const float* __restrict__ Hres,    // [4, 4]
    float* __restrict__ out)           // [B, 4, 4096]
{
    __shared__ __align__(16) float tile[NSTR][TSTR];   // ~64 KB, staged via async DMA
    __shared__ float sP[16];
    __shared__ float sHpre[NSTR];
    __shared__ float sHpost[NSTR];
    __shared__ float sSsq;

    const int t    = threadIdx.x;
    const int lane = t & 31;
    const int wv   = t >> 5;
    const int b    = blockIdx.x;
    const float* xb = x   + (size_t)b * (NSTR * CDIM);
    float*       ob = out + (size_t)b * (NSTR * CDIM);

    // ---- Per-block tiny state: gates + Sinkhorn-Knopp on 4x4 (thread 0, SALU-float) ----
    if (t == 0) {
        float P[16];
        #pragma unroll
        for (int i = 0; i < 16; ++i) P[i] = expf(Hres[i]);
        #pragma unroll
        for (int it = 0; it < 3; ++it) {
            #pragma unroll
            for (int i = 0; i < 4; ++i) {
                float s = P[4*i+0] + P[4*i+1] + P[4*i+2] + P[4*i+3] + EPSF;
                P[4*i+0] /= s; P[4*i+1] /= s; P[4*i+2] /= s; P[4*i+3] /= s;
            }
            #pragma unroll
            for (int j = 0; j < 4; ++j) {
                float s = P[j] + P[4+j] + P[8+j] + P[12+j] + EPSF;
                P[j] /= s; P[4+j] /= s; P[8+j] /= s; P[12+j] /= s;
            }
        }
        #pragma unroll
        for (int i = 0; i < 16; ++i) sP[i] = P[i];
        #pragma unroll
        for (int n = 0; n < NSTR; ++n) {
            sHpre[n]  = 1.0f / (1.0f + expf(-Hpre[n]));
            sHpost[n] = 2.0f / (1.0f + expf(-Hpost[n]));
        }
        sSsq = 0.0f;
    }

    // ---- Async DMA: x[b] (64 KB) global -> LDS (padded rows), ASYNCcnt path ----
    {
        const uint32_t lds0 = (uint32_t)(uintptr_t)&tile[0][0]; // LDS byte address
        #pragma unroll
        for (int n = 0; n < NSTR; ++n) {
            #pragma unroll
            for (int k = 0; k < 4; ++k) {
                const uint32_t off  = (uint32_t)((k * THREADS + t) * 16); // bytes in row
                const uint32_t ldsa = lds0 + (uint32_t)(n * TSTR * 4) + off;
                const uint32_t goff = (uint32_t)(n * CDIM * 4) + off;
                asm volatile("global_load_async_to_lds_b128 %0, %1, %2"
                             :: "v"(ldsa), "v"(goff), "s"(xb)
                             : "memory");
            }
        }
        asm volatile("s_wait_asynccnt 0" ::: "memory");
    }
    __syncthreads();

    const float hp0 = sHpre[0], hp1 = sHpre[1], hp2 = sHpre[2], hp3 = sHpre[3];

    // ---- Pass 1: x_agg = einsum('bnc,n->bc') on the matrix pipe ----
    // WMMA f32 16x16x4: A = sigmoid(H_pre) broadcast over 16 rows, B = 4x16 chunk of x.
    // A layout: lanes 0-15 hold K=0/1, lanes 16-31 hold K=2/3.
    // B layout: lane n holds column n; K=0/1 in lanes 0-15, K=2/3 in lanes 16-31.
    // D[m,n] = x_agg[c0+n] for every m -> read d[0]; halves duplicate, mask upper half.
    float ssq = 0.0f;
    {
        const int  hi    = (lane >= 16);
        const float lmask = hi ? 0.0f : 1.0f;          // cndmask, not EXEC (WMMA needs EXEC=all-1)
        v2f a;
        a.x = hi ? hp2 : hp0;
        a.y = hi ? hp3 : hp1;
        const int j0 = hi ? 2 : 0;
        const int cb = lane & 15;
        #pragma unroll 4
        for (int q = wv; q < CDIM / 16; q += 8) {
            const int cl = q * 16 + cb;
            v2f bm;
            bm.x = tile[j0 + 0][cl];
            bm.y = tile[j0 + 1][cl];
            v8f acc = {};
            acc = __builtin_amdgcn_wmma_f32_16x16x4_f32(
                    false, a, false, bm, (short)0, acc, false, false);
            const float g = bf16_roundtrip(acc[0]);
            ssq = fmaf(g * g, lmask, ssq);
        }
    }
    // wave32 reduction, then one LDS float atomic per wave
    #pragma unroll
    for (int m = 16; m > 0; m >>= 1) ssq += __shfl_xor(ssq, m, 32);
    if (lane == 0) atomicAdd(&sSsq, ssq);
    __syncthreads();

    const float rinv = 1.0f / sqrtf(sSsq * (1.0f / (float)CDIM) + EPSF);

    const float p00 = sP[0],  p01 = sP[1],  p02 = sP[2],  p03 = sP[3];
    const float p10 = sP[4],  p11 = sP[5],  p12 = sP[6],  p13 = sP[7];
    const float p20 = sP[8],  p21 = sP[9],  p22 = sP[10], p23 = sP[11];
    const float p30 = sP[12], p31 = sP[13], p32 = sP[14], p33 = sP[15];
    const float hq0 = sHpost[0], hq1 = sHpost[1], hq2 = sHpost[2], hq3 = sHpost[3];

    // ---- Pass 2: y_norm + 4x4 Sinkhorn mix, fat non-temporal B128 stores ----
    #pragma unroll
    for (int k = 0; k < 4; ++k) {
        const int c0 = 4 * t + 1024 * k;
        v4f a0 = *(const v4f*)&tile[0][c0];
        v4f a1 = *(const v4f*)&tile[1][c0];
        v4f a2 = *(const v4f*)&tile[2][c0];
        v4f a3 = *(const v4f*)&tile[3][c0];
        v4f g;
        g.x = bf16_roundtrip(hp0*a0.x + hp1*a1.x + hp2*a2.x + hp3*a3.x);
        g.y = bf16_roundtrip(hp0*a0.y + hp1*a1.y + hp2*a2.y + hp3*a3.y);
        g.z = bf16_roundtrip(hp0*a0.z + hp1*a1.z + hp2*a2.z + hp3*a3.z);
        g.w = bf16_roundtrip(hp0*a0.w + hp1*a1.w + hp2*a2.w + hp3*a3.w);
        v4f w4 = *(const v4f*)&w[c0];          // reused across all blocks -> L2 hit
        v4f yn = g * rinv * w4;

        v4f o;
        o = p00*a0 + p01*a1 + p02*a2 + p03*a3 + hq0*yn;
        __builtin_nontemporal_store(o, (v4f*)&ob[0*CDIM + c0]);
        o = p10*a0 + p11*a1 + p12*a2 + p13*a3 + hq1*yn;
        __builtin_nontemporal_store(o, (v4f*)&ob[1*CDIM + c0]);
        o = p20*a0 + p21*a1 + p22*a2 + p23*a3 + hq2*yn;
        __builtin_nontemporal_store(o, (v4f*)&ob[2*CDIM + c0]);
        o = p30*a0 + p31*a1 + p32*a2 + p33*a3 + hq3*yn;
        __builtin_nontemporal_store(o, (v4f*)&ob[3*CDIM + c0]);
    }
}

extern "C" void kernel_launch(void* const* d_in, const int* in_sizes, int n_in,
                              void* d_out, int out_size, void* d_ws, size_t ws_size,
                              hipStream_t stream) {
    (void)n_in; (void)out_size; (void)d_ws; (void)ws_size;
    const float* x     = (const float*)d_in[0];
    const float* w     = (const float*)d_in[1];
    const float* Hpre  = (const float*)d_in[2];
    const float* Hpost = (const float*)d_in[3];
    const float* Hres  = (const float*)d_in[4];
    float* out = (float*)d_out;

    const int B = in_sizes[0] / (NSTR * CDIM);   // 8192
    mhc_layer_kernel<<<B, THREADS, 0, stream>>>(x, w, Hpre, Hpost, Hres, out);
}